// _CovarianceQBatchNorm2d_65807488910045
// MI455X (gfx1250) — compile-verified
//
#include <hip/hip_runtime.h>

// CDNA5 / gfx1250, wave32. D = A(16x4) x B(4x16) + C via V_WMMA_F32_16X16X4_F32.
typedef __attribute__((ext_vector_type(2))) float v2f;
typedef __attribute__((ext_vector_type(8))) float v8f;

#define B_    32
#define C_    256
#define H_    64
#define W_    64
#define F_    64                  // C/4 quaternion channels
#define HW_   (H_ * W_)           // 4096
#define CHW_  (C_ * HW_)
#define FHW_  (F_ * HW_)          // stride between quaternion components
#define NTOT_ ((float)(B_ * HW_)) // samples per (component, channel)
#define EPS_  1e-5f

// workspace layout (floats)
#define PART_STRIDE 16                       // 14 used (4 sums + 10 products)
#define WS2_OFF     (F_ * B_ * PART_STRIDE)  // 32768 floats
#define AF_STRIDE   20                       // 16 (A row-major) + 4 (fused bias)

// ---------------------------------------------------------------------------
// Pass 1: per (channel f, batch b) partial sums S_q and cross-products P_qq'
// over the 4096 pixels of one image plane. Deterministic LDS tree reduction.
// ---------------------------------------------------------------------------
__global__ void __launch_bounds__(256) k_reduce(const float* __restrict__ x,
                                                float* __restrict__ ws) {
    const int f = blockIdx.x;   // channel
    const int b = blockIdx.y;   // batch image
    const int t = threadIdx.x;  // 0..255

    const size_t base = (size_t)b * CHW_ + (size_t)f * HW_;
    const float4* p0 = (const float4*)(x + base + 0 * (size_t)FHW_);
    const float4* p1 = (const float4*)(x + base + 1 * (size_t)FHW_);
    const float4* p2 = (const float4*)(x + base + 2 * (size_t)FHW_);
    const float4* p3 = (const float4*)(x + base + 3 * (size_t)FHW_);

    float s0 = 0.f, s1 = 0.f, s2 = 0.f, s3 = 0.f;
    float q00 = 0.f, q01 = 0.f, q02 = 0.f, q03 = 0.f, q11 = 0.f;
    float q12 = 0.f, q13 = 0.f, q22 = 0.f, q23 = 0.f, q33 = 0.f;

#pragma unroll
    for (int k = 0; k < 4; ++k) {
        const int idx = t + 256 * k;  // 1024 float4 per plane
        const float4 va = p0[idx];
        const float4 vb = p1[idx];
        const float4 vc = p2[idx];
        const float4 vd = p3[idx];
#define ACC(e)                                                     \
        { const float e0 = va.e, e1 = vb.e, e2 = vc.e, e3 = vd.e;  \
          s0 += e0; s1 += e1; s2 += e2; s3 += e3;                  \
          q00 += e0 * e0; q01 += e0 * e1; q02 += e0 * e2;          \
          q03 += e0 * e3; q11 += e1 * e1; q12 += e1 * e2;          \
          q13 += e1 * e3; q22 += e2 * e2; q23 += e2 * e3;          \
          q33 += e3 * e3; }
        ACC(x) ACC(y) ACC(z) ACC(w)
#undef ACC
    }

    __shared__ float red[256][14];
    float acc[14] = {s0, s1, s2, s3, q00, q01, q02, q03, q11, q12, q13, q22, q23, q33};
#pragma unroll
    for (int j = 0; j < 14; ++j) red[t][j] = acc[j];
    __syncthreads();
    for (int s = 128; s > 0; s >>= 1) {
        if (t < s) {
#pragma unroll
            for (int j = 0; j < 14; ++j) red[t][j] += red[t + s][j];
        }
        __syncthreads();
    }
    if (t == 0) {
        float* dst = ws + (size_t)(f * B_ + b) * PART_STRIDE;
#pragma unroll
        for (int j = 0; j < 14; ++j) dst[j] = red[0][j];
    }
}

// ---------------------------------------------------------------------------
// Pass 2: per channel f -> mean, cov(+eps I), 4x4 Cholesky, L^-1,
// A = W_f * L^-1 (4x4), cfused = bias_f - A*mean. One thread per channel.
// ---------------------------------------------------------------------------
__global__ void k_finalize(const float* __restrict__ weight,
                           const float* __restrict__ bias,
                           float* __restrict__ ws) {
    const int f = threadIdx.x;
    if (f >= F_) return;

    float acc[14];
#pragma unroll
    for (int j = 0; j < 14; ++j) acc[j] = 0.f;
    for (int p = 0; p < B_; ++p) {
        const float* src = ws + (size_t)(f * B_ + p) * PART_STRIDE;
#pragma unroll
        for (int j = 0; j < 14; ++j) acc[j] += src[j];
    }
    const float invN = 1.0f / NTOT_;
    const float m0 = acc[0] * invN, m1 = acc[1] * invN;
    const float m2 = acc[2] * invN, m3 = acc[3] * invN;
    const float c00 = acc[4] * invN - m0 * m0 + EPS_;
    const float c01 = acc[5] * invN - m0 * m1;
    const float c02 = acc[6] * invN - m0 * m2;
    const float c03 = acc[7] * invN - m0 * m3;
    const float c11 = acc[8] * invN - m1 * m1 + EPS_;
    const float c12 = acc[9] * invN - m1 * m2;
    const float c13 = acc[10] * invN - m1 * m3;
    const float c22 = acc[11] * invN - m2 * m2 + EPS_;
    const float c23 = acc[12] * invN - m2 * m3;
    const float c33 = acc[13] * invN - m3 * m3 + EPS_;

    // Cholesky (lower)
    const float L00 = sqrtf(c00);
    const float L10 = c01 / L00, L20 = c02 / L00, L30 = c03 / L00;
    const float L11 = sqrtf(c11 - L10 * L10);
    const float L21 = (c12 - L20 * L10) / L11;
    const float L31 = (c13 - L30 * L10) / L11;
    const float L22 = sqrtf(c22 - L20 * L20 - L21 * L21);
    const float L32 = (c23 - L30 * L20 - L31 * L21) / L22;
    const float L33 = sqrtf(c33 - L30 * L30 - L31 * L31 - L32 * L32);

    // M = L^-1 (lower triangular)
    const float M00 = 1.f / L00, M11 = 1.f / L11, M22 = 1.f / L22, M33 = 1.f / L33;
    const float M10 = -M11 * (L10 * M00);
    const float M20 = -M22 * (L20 * M00 + L21 * M10);
    const float M21 = -M22 * (L21 * M11);
    const float M30 = -M33 * (L30 * M00 + L31 * M10 + L32 * M20);
    const float M31 = -M33 * (L31 * M11 + L32 * M21);
    const float M32 = -M33 * (L32 * M22);
    const float M[4][4] = {{M00, 0.f, 0.f, 0.f},
                           {M10, M11, 0.f, 0.f},
                           {M20, M21, M22, 0.f},
                           {M30, M31, M32, M33}};
    const float mean[4] = {m0, m1, m2, m3};

    float* dst = ws + WS2_OFF + (size_t)f * AF_STRIDE;
#pragma unroll
    for (int i = 0; i < 4; ++i) {
        float Ai[4];
#pragma unroll
        for (int j = 0; j < 4; ++j) {
            float s = 0.f;
            for (int k = j; k < 4; ++k) s += weight[(i * 4 + k) * F_ + f] * M[k][j];
            Ai[j] = s;
            dst[i * 4 + j] = s;
        }
        float cb = bias[i * F_ + f];
#pragma unroll
        for (int j = 0; j < 4; ++j) cb -= Ai[j] * mean[j];
        dst[16 + i] = cb;
    }
}

// ---------------------------------------------------------------------------
// Pass 3: streaming apply, out = A*x + c per 16-pixel tile via
// v_wmma_f32_16x16x4_f32. A-operand: 4x4 matrix replicated over the 16 M-rows
// (Arep[m,k] = A[m%4,k]); B-operand: 4 components x 16 pixels; C carries the
// fused bias. D VGPR r, lanes 0-15 = component r for the 16 pixels.
// Manually unrolled x4: 8 loads in flight per wave before the first wait,
// one predicated store region per 4 tiles (exec save/restore amortized).
// ---------------------------------------------------------------------------
__global__ void __launch_bounds__(256) k_apply(const float* __restrict__ x,
                                               const float* __restrict__ ws,
                                               float* __restrict__ out) {
    const int f = blockIdx.x;
    const int b = blockIdx.y;
    const int lane = threadIdx.x & 31;
    const int wave = threadIdx.x >> 5;     // 8 waves/block
    const int half = lane >> 4;            // 0: lanes 0-15, 1: lanes 16-31
    const int col  = lane & 15;            // N (pixel within tile)
    const int row  = col & 3;              // component for this M-row

    // A fragment: v0 <- K = 2*half, v1 <- K = 2*half+1 (ISA 32-bit 16x4 layout)
    const float* Af = ws + WS2_OFF + (size_t)f * AF_STRIDE;
    v2f a;
    a.x = Af[row * 4 + 2 * half + 0];
    a.y = Af[row * 4 + 2 * half + 1];

    // C fragment: VGPR r holds rows {r, r+8} -> component r%4 either way.
    v8f cfrag;
#pragma unroll
    for (int r = 0; r < 8; ++r) cfrag[r] = Af[16 + (r & 3)];

    const size_t base = (size_t)b * CHW_ + (size_t)f * HW_;
    // B fragment sources: components (2*half) and (2*half+1)
    const float* px0 = x + base + (size_t)(2 * half) * FHW_;
    const float* px1 = px0 + FHW_;
    float* o0 = out + base;
    float* o1 = o0 + FHW_;
    float* o2 = o1 + FHW_;
    float* o3 = o2 + FHW_;

    const int t0 = wave * 32;              // 256 tiles per (f,b) plane
    for (int tile = t0; tile < t0 + 32; tile += 4) {
        const int p0i = (tile + 0) * 16 + col;
        const int p1i = (tile + 1) * 16 + col;
        const int p2i = (tile + 2) * 16 + col;
        const int p3i = (tile + 3) * 16 + col;

        // Issue all 8 loads before any consumer -> deep MLP per wave.
        v2f bf0, bf1, bf2, bf3;
        bf0.x = px0[p0i]; bf0.y = px1[p0i];
        bf1.x = px0[p1i]; bf1.y = px1[p1i];
        bf2.x = px0[p2i]; bf2.y = px1[p2i];
        bf3.x = px0[p3i]; bf3.y = px1[p3i];

        // (neg_a, A, neg_b, B, c_mod, C, reuse_a, reuse_b)
        const v8f d0 = __builtin_amdgcn_wmma_f32_16x16x4_f32(
            false, a, false, bf0, (short)0, cfrag, false, false);
        const v8f d1 = __builtin_amdgcn_wmma_f32_16x16x4_f32(
            false, a, false, bf1, (short)0, cfrag, false, false);
        const v8f d2 = __builtin_amdgcn_wmma_f32_16x16x4_f32(
            false, a, false, bf2, (short)0, cfrag, false, false);
        const v8f d3 = __builtin_amdgcn_wmma_f32_16x16x4_f32(
            false, a, false, bf3, (short)0, cfrag, false, false);

        if (half == 0) {                   // rows 4-15 of D duplicate rows 0-3
            o0[p0i] = d0[0]; o1[p0i] = d0[1]; o2[p0i] = d0[2]; o3[p0i] = d0[3];
            o0[p1i] = d1[0]; o1[p1i] = d1[1]; o2[p1i] = d1[2]; o3[p1i] = d1[3];
            o0[p2i] = d2[0]; o1[p2i] = d2[1]; o2[p2i] = d2[2]; o3[p2i] = d2[3];
            o0[p3i] = d3[0]; o1[p3i] = d3[1]; o2[p3i] = d3[2]; o3[p3i] = d3[3];
        }
    }
}

extern "C" void kernel_launch(void* const* d_in, const int* in_sizes, int n_in,
                              void* d_out, int out_size, void* d_ws, size_t ws_size,
                              hipStream_t stream) {
    const float* x      = (const float*)d_in[0];   // [32,256,64,64]
    const float* weight = (const float*)d_in[1];   // [4,4,64]
    const float* bias   = (const float*)d_in[2];   // [4,64]
    float* out = (float*)d_out;
    float* ws  = (float*)d_ws;                     // ~136 KiB used

    dim3 grid(F_, B_);
    k_reduce  <<<grid, 256, 0, stream>>>(x, ws);
    k_finalize<<<1,    64,  0, stream>>>(weight, bias, ws);
    k_apply   <<<grid, 256, 0, stream>>>(x, ws, out);
}